// MultiheadAttention_44753559224445
// MI455X (gfx1250) — compile-verified
//
#include <hip/hip_runtime.h>

#define BQ     4
#define SEQ    2048
#define DIM    512
#define NHEAD  8
#define HD     64
#define SCALE_Q 0.125f   // HD^-0.5

typedef float v2f __attribute__((ext_vector_type(2)));
typedef float v4f __attribute__((ext_vector_type(4)));
typedef float v8f __attribute__((ext_vector_type(8)));

// ---------------------------------------------------------------------------
// GEMM: C[M,N] = (X[M,K] @ W[N,K]^T + bias[N]) * scale      (nn.Linear style)
// 256 threads = 8 waves; block tile 128(M) x 64(N); K staged in 32-chunks.
// Per wave: 16x64 C tile = 4 x v_wmma_f32_16x16x4_f32 accumulators.
// X is read-once -> NT loads. NT_STORE=true for the final output projection
// (write-once to d_out); false for Q/K/V/O workspace (re-read, keep in L2).
// ---------------------------------------------------------------------------
template <bool NT_STORE>
__global__ __launch_bounds__(256)
void proj_gemm(const float* __restrict__ X, const float* __restrict__ W,
               const float* __restrict__ bias, float* __restrict__ C,
               int M, int N, int K, float scale)
{
    __shared__ float Xs[128][33];   // +1 pad: kills bank conflicts on A-frag reads
    __shared__ float Ws[64][33];

    const int t    = threadIdx.x;
    const int wave = t >> 5;
    const int lane = t & 31;
    const int half = lane >> 4;     // 0: K{0,1}/rows 0-7 ; 1: K{2,3}/rows 8-15
    const int l16  = lane & 15;

    const int nb = N / 64;
    const int m0 = (blockIdx.x / nb) * 128;
    const int n0 = (blockIdx.x % nb) * 64;

    v8f acc[4] = {};

    for (int k0 = 0; k0 < K; k0 += 32) {
        // stage X tile 128x32 (coalesced b128 NT loads: streamed, read-once)
        #pragma unroll
        for (int i = 0; i < 4; ++i) {
            int lin = (i * 256 + t) * 4;            // 0..4095
            int r = lin >> 5, c = lin & 31;
            v4f v = __builtin_nontemporal_load(
                (const v4f*)&X[(size_t)(m0 + r) * K + k0 + c]);
            Xs[r][c] = v.x; Xs[r][c+1] = v.y; Xs[r][c+2] = v.z; Xs[r][c+3] = v.w;
        }
        // stage W tile 64x32 (rows n0..n0+63 of W; reused across M -> RT)
        #pragma unroll
        for (int i = 0; i < 2; ++i) {
            int lin = (i * 256 + t) * 4;            // 0..2047
            int r = lin >> 5, c = lin & 31;
            v4f v = *(const v4f*)&W[(size_t)(n0 + r) * K + k0 + c];
            Ws[r][c] = v.x; Ws[r][c+1] = v.y; Ws[r][c+2] = v.z; Ws[r][c+3] = v.w;
        }
        __syncthreads();

        const int arow = wave * 16 + l16;           // A: lane -> M row
        #pragma unroll
        for (int kk = 0; kk < 32; kk += 4) {
            const int kb = kk + half * 2;           // A/B: half -> K pair
            v2f a;
            a.x = Xs[arow][kb];
            a.y = Xs[arow][kb + 1];
            #pragma unroll
            for (int s = 0; s < 4; ++s) {
                v2f bf;                              // B: lane -> N col
                bf.x = Ws[s * 16 + l16][kb];
                bf.y = Ws[s * 16 + l16][kb + 1];
                acc[s] = __builtin_amdgcn_wmma_f32_16x16x4_f32(
                    false, a, false, bf, (short)0, acc[s], false, false);
            }
        }
        __syncthreads();
    }

    // epilogue: (acc + bias) * scale, C layout: lane=col, vgpr i = row i+8*half
    #pragma unroll
    for (int s = 0; s < 4; ++s) {
        const int n  = n0 + s * 16 + l16;
        const float bv = bias[n];
        #pragma unroll
        for (int i = 0; i < 8; ++i) {
            const int r = m0 + wave * 16 + i + half * 8;
            const float cv = (acc[s][i] + bv) * scale;
            if (NT_STORE)
                __builtin_nontemporal_store(cv, &C[(size_t)r * N + n]);
            else
                C[(size_t)r * N + n] = cv;
        }
    }
}

// ---------------------------------------------------------------------------
// Attention: one block = one (b, q-tile of 16) for ALL 8 heads (wave = head).
// Pass 1: flash-style running row-max/row-sum over causal k-tiles.
// Pass 2: recompute scores, normalize, assemble [16q][16k][8n] probabilities
//         in LDS, write attn with coalesced NT b128 stores, accumulate P@V.
// Tail  : zero-fill the causal upper triangle (reference softmax -> exact 0).
// attn is 512MB write-once -> NT stores keep Q/K/V (48MB) resident in L2.
// ---------------------------------------------------------------------------
__global__ __launch_bounds__(256)
void attn_kernel(const float* __restrict__ Qw, const float* __restrict__ Kw,
                 const float* __restrict__ Vw,
                 const unsigned char* __restrict__ keymask,
                 float* __restrict__ attn, float* __restrict__ O)
{
    __shared__ __attribute__((aligned(16))) float Qs[NHEAD][16][65]; // pad 65
    __shared__ __attribute__((aligned(16))) float Pbuf[16 * 132];    // pad 132

    const int t    = threadIdx.x;
    const int n    = t >> 5;          // head index == wave index
    const int lane = t & 31;
    const int half = lane >> 4;
    const int l16  = lane & 15;

    const int b  = blockIdx.x >> 7;           // / (SEQ/16)
    const int q0 = (blockIdx.x & 127) * 16;

    // stage the q-tile for all heads: 16 rows x 512 cols
    const size_t rowQ = ((size_t)b * SEQ + q0) * DIM;
    #pragma unroll
    for (int i = 0; i < 8; ++i) {
        int lin = (i * 256 + t) * 4;          // 0..8191
        int hh  = lin >> 10;
        int rem = lin & 1023;
        int m = rem >> 6, c = rem & 63;
        v4f v = *(const v4f*)&Qw[rowQ + (size_t)m * DIM + hh * HD + c];
        Qs[hh][m][c] = v.x; Qs[hh][m][c+1] = v.y;
        Qs[hh][m][c+2] = v.z; Qs[hh][m][c+3] = v.w;
    }
    __syncthreads();

    const int    ktiles = (q0 >> 4) + 1;                // causal extent
    const size_t kvHead = (size_t)b * SEQ * DIM + (size_t)n * HD;

    float mrow[8], lrow[8];
    #pragma unroll
    for (int i = 0; i < 8; ++i) { mrow[i] = -1e30f; lrow[i] = 0.f; }

    // ---------------- pass 1: row max / row sum -----------------------------
    for (int kt = 0; kt < ktiles; ++kt) {
        const int k0 = kt * 16;
        if (kt + 1 < ktiles)   // prefetch next K tile row for this lane
            __builtin_prefetch(&Kw[kvHead + (size_t)(k0 + 16 + l16) * DIM], 0, 1);
        v8f s = {};
        #pragma unroll
        for (int kk = 0; kk < HD; kk += 4) {
            const int kb = kk + half * 2;
            v2f a;  a.x = Qs[n][l16][kb];  a.y = Qs[n][l16][kb + 1];
            float2 kv = *(const float2*)&Kw[kvHead + (size_t)(k0 + l16) * DIM + kb];
            v2f bf; bf.x = kv.x; bf.y = kv.y;
            s = __builtin_amdgcn_wmma_f32_16x16x4_f32(
                false, a, false, bf, (short)0, s, false, false);
        }
        const int  kc   = k0 + l16;                       // this lane's k col
        const bool kmok = keymask[(size_t)b * SEQ + kc] != 0;
        #pragma unroll
        for (int i = 0; i < 8; ++i) {
            const int qr = q0 + i + half * 8;
            float v = s[i];
            if (kc > qr || !kmok) v = -1e30f;             // causal + key mask
            float tm = v;                                  // row-max over 16 cols
            tm = fmaxf(tm, __shfl_xor(tm, 1));
            tm = fmaxf(tm, __shfl_xor(tm, 2));
            tm = fmaxf(tm, __shfl_xor(tm, 4));
            tm = fmaxf(tm, __shfl_xor(tm, 8));
            const float mnew = fmaxf(mrow[i], tm);
            float p = __expf(v - mnew);
            p += __shfl_xor(p, 1);
            p += __shfl_xor(p, 2);
            p += __shfl_xor(p, 4);
            p += __shfl_xor(p, 8);
            lrow[i] = lrow[i] * __expf(mrow[i] - mnew) + p;
            mrow[i] = mnew;
        }
    }
    float rl[8];
    #pragma unroll
    for (int i = 0; i < 8; ++i) rl[i] = 1.0f / lrow[i];

    // ---------------- pass 2: normalize, emit attn, accumulate P@V ----------
    v8f oacc[4] = {};
    for (int kt = 0; kt < ktiles; ++kt) {
        const int k0 = kt * 16;
        if (kt + 1 < ktiles) { // prefetch next K/V tile rows for this lane
            __builtin_prefetch(&Kw[kvHead + (size_t)(k0 + 16 + l16) * DIM], 0, 1);
            __builtin_prefetch(&Vw[kvHead + (size_t)(k0 + 16 + l16) * DIM], 0, 1);
        }
        v8f s = {};
        #pragma unroll
        for (int kk = 0; kk < HD; kk += 4) {
            const int kb = kk + half * 2;
            v2f a;  a.x = Qs[n][l16][kb];  a.y = Qs[n][l16][kb + 1];
            float2 kv = *(const float2*)&Kw[kvHead + (size_t)(k0 + l16) * DIM + kb];
            v2f bf; bf.x = kv.x; bf.y = kv.y;
            s = __builtin_amdgcn_wmma_f32_16x16x4_f32(
                false, a, false, bf, (short)0, s, false, false);
        }
        const int  kc   = k0 + l16;
        const bool kmok = keymask[(size_t)b * SEQ + kc] != 0;
        #pragma unroll
        for (int i = 0; i < 8; ++i) {
            const int qr = q0 + i + half * 8;
            float v = s[i];
            if (kc > qr || !kmok) v = -1e30f;
            const float p = __expf(v - mrow[i]) * rl[i];
            Pbuf[(i + half * 8) * 132 + l16 * 8 + n] = p;   // [q][k][n]
        }
        __syncthreads();

        // coalesced attn write: 512B contiguous per q-row ([k][n] innermost)
        #pragma unroll
        for (int rep = 0; rep < 2; ++rep) {
            const int e = (rep * 256 + t) * 4;              // 0..2047
            const int q = e >> 7, off = e & 127;
            v4f v = *(const v4f*)&Pbuf[q * 132 + off];
            __builtin_nontemporal_store(v,
                (v4f*)&attn[(((size_t)b * SEQ + q0 + q) * SEQ + k0) * NHEAD + off]);
        }

        // O tile += P[16x16] @ V[16x64]
        #pragma unroll
        for (int kk = 0; kk < 16; kk += 4) {
            const int kb = kk + half * 2;
            v2f a;                                           // A = P, lane = q row
            a.x = Pbuf[l16 * 132 + kb * 8 + n];
            a.y = Pbuf[l16 * 132 + (kb + 1) * 8 + n];
            #pragma unroll
            for (int ss = 0; ss < 4; ++ss) {
                const float* vp = &Vw[kvHead + (size_t)(k0 + kb) * DIM + ss * 16 + l16];
                v2f bf; bf.x = vp[0]; bf.y = vp[DIM];        // B = V, lane = h col
                oacc[ss] = __builtin_amdgcn_wmma_f32_16x16x4_f32(
                    false, a, false, bf, (short)0, oacc[ss], false, false);
            }
        }
        __syncthreads();
    }

    // ---------------- zero-fill causal upper triangle (NT, write-once) ------
    const v4f z = {0.f, 0.f, 0.f, 0.f};
    for (int kt = ktiles; kt < SEQ / 16; ++kt) {
        const int k0 = kt * 16;
        #pragma unroll
        for (int rep = 0; rep < 2; ++rep) {
            const int e = (rep * 256 + t) * 4;
            const int q = e >> 7, off = e & 127;
            __builtin_nontemporal_store(z,
                (v4f*)&attn[(((size_t)b * SEQ + q0 + q) * SEQ + k0) * NHEAD + off]);
        }
    }

    // ---------------- store O tile (re-read by out-projection -> RT) --------
    const size_t orow = ((size_t)b * SEQ + q0) * DIM + (size_t)n * HD;
    #pragma unroll
    for (int ss = 0; ss < 4; ++ss)
        #pragma unroll
        for (int i = 0; i < 8; ++i)
            O[orow + (size_t)(i + half * 8) * DIM + ss * 16 + l16] = oacc[ss][i];
}

// ---------------------------------------------------------------------------
extern "C" void kernel_launch(void* const* d_in, const int* in_sizes, int n_in,
                              void* d_out, int out_size, void* d_ws, size_t ws_size,
                              hipStream_t stream)
{
    const float* query = (const float*)d_in[0];
    const float* key   = (const float*)d_in[1];
    const float* value = (const float*)d_in[2];
    const unsigned char* keymask = (const unsigned char*)d_in[3];
    // d_in[4] = attn_mask (causal) -> applied analytically
    const float* Wq = (const float*)d_in[5];
    const float* bq = (const float*)d_in[6];
    const float* Wk = (const float*)d_in[7];
    const float* bk = (const float*)d_in[8];
    const float* Wv = (const float*)d_in[9];
    const float* bv = (const float*)d_in[10];
    const float* Wo = (const float*)d_in[11];
    const float* bo = (const float*)d_in[12];

    const size_t tok = (size_t)BQ * SEQ * DIM;   // 4,194,304 floats
    float* Qw = (float*)d_ws;
    float* Kw = Qw + tok;
    float* Vw = Kw + tok;
    float* Ow = Vw + tok;

    float* out_p  = (float*)d_out;               // [B,S,D]
    float* attn_p = out_p + tok;                 // [B,S,S,NH]

    const int M = BQ * SEQ;                      // 8192
    const dim3 gGemm((M / 128) * (DIM / 64));    // 512 blocks
    const dim3 blk(256);

    proj_gemm<false><<<gGemm, blk, 0, stream>>>(query, Wq, bq, Qw, M, DIM, DIM, SCALE_Q);
    proj_gemm<false><<<gGemm, blk, 0, stream>>>(key,   Wk, bk, Kw, M, DIM, DIM, 1.0f);
    proj_gemm<false><<<gGemm, blk, 0, stream>>>(value, Wv, bv, Vw, M, DIM, DIM, 1.0f);

    attn_kernel<<<dim3(BQ * (SEQ / 16)), blk, 0, stream>>>(Qw, Kw, Vw, keymask,
                                                           attn_p, Ow);

    proj_gemm<true><<<gGemm, blk, 0, stream>>>(Ow, Wo, bo, out_p, M, DIM, DIM, 1.0f);
}